// RwkvModel_46050639348061
// MI455X (gfx1250) — compile-verified
//
#include <hip/hip_runtime.h>
#include <math.h>

// ---------------- model constants (from reference) ----------------
#define L_    4
#define C_    64
#define N_    64
#define F_    256
#define T_    256
#define B_    32
#define DIN   16
#define DOUT  10

#define EPS_LN 1e-5f
#define EPS_GN 6.4e-4f          // 1e-5 * 8^2 per RWKV-v5 head_size_divisor

// ---------------- launch shape ----------------
#define ROWS     2               // batch rows per workgroup (WMMA M-tile padded to 16)
#define NBLK     (B_ / ROWS)     // 16 workgroups, one chain each
#define NTHREADS 256
#define NWAVES   8               // wave32

typedef float v2f __attribute__((ext_vector_type(2)));
typedef float v8f __attribute__((ext_vector_type(8)));

// ---- wave32 all-lane sum via shuffle (LN/GN over 64 channels, 2 per lane) ----
__device__ __forceinline__ float wred_sum(float v) {
#pragma unroll
  for (int m = 16; m >= 1; m >>= 1) v += __shfl_xor(v, m, 32);
  return v;
}

// LayerNorm for a row laid out as (col=lane, col=lane+32) in one wave.
__device__ __forceinline__ void ln_pair(float v0, float v1, float eps,
                                        float w0, float w1, float b0, float b1,
                                        float& o0, float& o1) {
  const float mean = wred_sum(v0 + v1) * (1.f / 64.f);
  const float d0 = v0 - mean, d1 = v1 - mean;
  const float var = wred_sum(d0 * d0 + d1 * d1) * (1.f / 64.f);
  const float rs = rsqrtf(var + eps);
  o0 = d0 * rs * w0 + b0;
  o1 = d1 * rs * w1 + b1;
}

// ---------------- one 16x16 output tile of  Out = A @ W^T  -------------------
// A  : LDS  [16][K]   fp32 (rows >= ROWS are zero)
// W  : glob [NOUT][K] fp32 (so B[k][n] = W[n][k])
// Out: LDS  [16][NOUT] ; rows >= ROWS forced to 0 so Out may later feed as A.
// v_wmma_f32_16x16x4_f32 operand layout (ISA 7.12.2):
//   A: lane m (m<16)->rows, vgpr0/1 = K {0,1}, lanes 16..31 = K {2,3}
//   B: row k striped across lanes within a VGPR (vgpr r = rows {r, r+2})
//   D: vgpr r = row r (lanes 0..15) / row r+8 (lanes 16..31)
// ACT: 0 none, 1 silu, 2 sigmoid, 3 relu^2 (fused at store)
template <int ACT, int K, int NOUT>
__device__ __forceinline__ void mm_tile(const float* As, const float* __restrict__ W,
                                        float* Out, int tile, int lane) {
  const int m  = lane & 15;
  const int kb = (lane >> 4) << 1;           // 0 or 2
  const int n  = tile * 16 + m;
  const float* wrow = W + n * K + kb;
  const float* arow = As + m * K + kb;
  v8f c = {};
#pragma unroll
  for (int k0 = 0; k0 < K; k0 += 4) {
    v2f a; a.x = arow[k0]; a.y = arow[k0 + 1];
    v2f b; b.x = wrow[k0]; b.y = wrow[k0 + 1];
    c = __builtin_amdgcn_wmma_f32_16x16x4_f32(false, a, false, b, (short)0, c,
                                              false, false);
  }
  const int rbase = (lane < 16) ? 0 : 8;
#pragma unroll
  for (int r = 0; r < 8; ++r) {
    const int row = rbase + r;
    float x = c[r];
    if (ACT == 1)      x = x / (1.f + __expf(-x));            // silu
    else if (ACT == 2) x = 1.f / (1.f + __expf(-x));          // sigmoid
    else if (ACT == 3) { x = fmaxf(x, 0.f); x *= x; }         // relu^2
    Out[row * NOUT + n] = (row < ROWS) ? x : 0.f;             // keep pad rows 0
  }
}

// ---------------- shared memory layout (190,464 bytes, < 320KB/WGP) ----------
#define SM_STATE  0                                   // [L][ROWS][64][64]
#define SM_H      (SM_STATE + L_ * ROWS * N_ * N_)    // [16][64]
#define SM_A0     (SM_H  + 16 * C_)                   // zero-padded A buffers
#define SM_A1     (SM_A0 + 16 * C_)
#define SM_A2     (SM_A1 + 16 * C_)
#define SM_A3     (SM_A2 + 16 * C_)
#define SM_AK     (SM_A3 + 16 * C_)                   // [16][256]
#define SM_M0     (SM_AK + 16 * F_)                   // matmul outputs
#define SM_M1     (SM_M0 + 16 * C_)
#define SM_M2     (SM_M1 + 16 * C_)
#define SM_M3     (SM_M2 + 16 * C_)
#define SM_ATTXX  (SM_M3 + 16 * C_)                   // [L][ROWS][64]
#define SM_FFNXX  (SM_ATTXX + L_ * ROWS * C_)
#define SM_UL     (SM_FFNXX + L_ * ROWS * C_)         // time_first   [L][64]
#define SM_WL     (SM_UL + L_ * N_)                   // exp(-exp(td))[L][64]
#define SM_FLOATS (SM_WL + L_ * N_)
#define SMEM_BYTES (SM_FLOATS * sizeof(float))

__global__ void __launch_bounds__(NTHREADS, 1)
rwkv5_forward(const float* __restrict__ x,
              const float* __restrict__ enc_w,  const float* __restrict__ enc_b,
              const float* __restrict__ ln0_w,  const float* __restrict__ ln0_b,
              const float* __restrict__ lno_w,  const float* __restrict__ lno_b,
              const float* __restrict__ head_w, const float* __restrict__ head_b,
              const float* __restrict__ ln1_w,  const float* __restrict__ ln1_b,
              const float* __restrict__ ln2_w,  const float* __restrict__ ln2_b,
              const float* __restrict__ lnx_w,  const float* __restrict__ lnx_b,
              const float* __restrict__ mix_k,  const float* __restrict__ mix_v,
              const float* __restrict__ mix_r,  const float* __restrict__ mix_g,
              const float* __restrict__ Wr, const float* __restrict__ Wk,
              const float* __restrict__ Wv, const float* __restrict__ Wg,
              const float* __restrict__ Wo,
              const float* __restrict__ time_decay, const float* __restrict__ time_first,
              const float* __restrict__ fmix_k, const float* __restrict__ fmix_r,
              const float* __restrict__ fWk, const float* __restrict__ fWv,
              const float* __restrict__ fWr,
              float* __restrict__ out) {
  extern __shared__ float smem[];
  float* state = smem + SM_STATE;
  float* hbuf  = smem + SM_H;
  float* a0 = smem + SM_A0; float* a1 = smem + SM_A1;
  float* a2 = smem + SM_A2; float* a3 = smem + SM_A3;
  float* aK = smem + SM_AK;
  float* m0 = smem + SM_M0; float* m1 = smem + SM_M1;
  float* m2 = smem + SM_M2; float* m3 = smem + SM_M3;
  float* attxx = smem + SM_ATTXX;
  float* ffnxx = smem + SM_FFNXX;
  float* uL = smem + SM_UL;
  float* wL = smem + SM_WL;

  const int tid  = threadIdx.x;
  const int wave = tid >> 5;
  const int lane = tid & 31;
  const int b0   = blockIdx.x * ROWS;     // batch rows [b0, b0+ROWS)

  // ---------------- init: zero everything we carry / pad rows ----------------
  for (int i = tid; i < L_ * ROWS * N_ * N_; i += NTHREADS) state[i] = 0.f;
  for (int i = tid; i < 16 * C_; i += NTHREADS) {
    hbuf[i] = 0.f; a0[i] = 0.f; a1[i] = 0.f; a2[i] = 0.f; a3[i] = 0.f;
    m0[i] = 0.f; m1[i] = 0.f; m2[i] = 0.f; m3[i] = 0.f;
  }
  for (int i = tid; i < 16 * F_; i += NTHREADS) aK[i] = 0.f;
  for (int i = tid; i < L_ * ROWS * C_; i += NTHREADS) { attxx[i] = 0.f; ffnxx[i] = 0.f; }
  for (int i = tid; i < L_ * N_; i += NTHREADS) {
    wL[i] = __expf(-__expf(time_decay[i]));   // per-channel decay
    uL[i] = time_first[i];
  }
  __syncthreads();

  // ---------------- sequential scan over time ----------------
  for (int t = 0; t < T_; ++t) {
    // ---- encode + ln0 : h = LN(x_t @ enc_w^T + enc_b) ----
    if (wave < ROWS) {
      const int row = wave;
      const float* xp = x + ((size_t)(b0 + row) * T_ + t) * DIN;   // x[B,1,T,Din]
      float h0 = enc_b[lane], h1 = enc_b[lane + 32];
#pragma unroll
      for (int d = 0; d < DIN; ++d) {
        const float xv = xp[d];
        h0 = fmaf(xv, enc_w[lane * DIN + d], h0);
        h1 = fmaf(xv, enc_w[(lane + 32) * DIN + d], h1);
      }
      float n0, n1;
      ln_pair(h0, h1, EPS_LN, ln0_w[lane], ln0_w[lane + 32],
              ln0_b[lane], ln0_b[lane + 32], n0, n1);
      hbuf[row * C_ + lane] = n0;
      hbuf[row * C_ + lane + 32] = n1;
    }

    for (int l = 0; l < L_; ++l) {
      const float* pWr = Wr + l * C_ * C_;  const float* pWk = Wk + l * C_ * C_;
      const float* pWv = Wv + l * C_ * C_;  const float* pWg = Wg + l * C_ * C_;
      const float* pWo = Wo + l * C_ * C_;
      const float* pfWr = fWr + l * C_ * C_;
      const float* pfWk = fWk + l * F_ * C_;
      const float* pfWv = fWv + l * C_ * F_;

      // ---- phase 1: ln1 + time-mix -> a0..a3 (xk,xv,xr,xg); update att_xx ----
      if (wave < ROWS) {
        const int row = wave, i0 = row * C_ + lane;
        float n0, n1;
        ln_pair(hbuf[i0], hbuf[i0 + 32], EPS_LN,
                ln1_w[l * C_ + lane], ln1_w[l * C_ + lane + 32],
                ln1_b[l * C_ + lane], ln1_b[l * C_ + lane + 32], n0, n1);
        const int xi = (l * ROWS + row) * C_ + lane;
        const float p0 = attxx[xi], p1 = attxx[xi + 32];
        attxx[xi] = n0; attxx[xi + 32] = n1;
        const float mk0 = mix_k[l * C_ + lane], mk1 = mix_k[l * C_ + lane + 32];
        const float mv0 = mix_v[l * C_ + lane], mv1 = mix_v[l * C_ + lane + 32];
        const float mr0 = mix_r[l * C_ + lane], mr1 = mix_r[l * C_ + lane + 32];
        const float mg0 = mix_g[l * C_ + lane], mg1 = mix_g[l * C_ + lane + 32];
        a0[i0] = n0 * mk0 + p0 * (1.f - mk0);  a0[i0 + 32] = n1 * mk1 + p1 * (1.f - mk1);
        a1[i0] = n0 * mv0 + p0 * (1.f - mv0);  a1[i0 + 32] = n1 * mv1 + p1 * (1.f - mv1);
        a2[i0] = n0 * mr0 + p0 * (1.f - mr0);  a2[i0 + 32] = n1 * mr1 + p1 * (1.f - mr1);
        a3[i0] = n0 * mg0 + p0 * (1.f - mg0);  a3[i0 + 32] = n1 * mg1 + p1 * (1.f - mg1);
      } else if (t == 0) {
        // idle waves warm L2 with this layer's weights once (global_prefetch_b8)
        const char* ws[3] = {(const char*)pWr, (const char*)pfWk, (const char*)pfWv};
        const int   nb[3] = {5 * C_ * C_ * 4, F_ * C_ * 4, C_ * F_ * 4}; // Wr..Wo contiguous? prefetch each
        for (int a = 0; a < 3; ++a)
          for (int off = ((wave - ROWS) * 32 + lane) * 128; off < nb[a];
               off += (NWAVES - ROWS) * 32 * 128)
            __builtin_prefetch(ws[a] + off, 0, 3);
      }
      __syncthreads();

      // ---- phase 2: r,k,v,g = WMMA matmuls (16 tile-jobs over 8 waves) ----
      for (int job = wave; job < 16; job += NWAVES) {
        const int which = job >> 2, tile = job & 3;
        if      (which == 0) mm_tile<0, C_, C_>(a2, pWr, m0, tile, lane);   // r
        else if (which == 1) mm_tile<0, C_, C_>(a0, pWk, m1, tile, lane);   // k
        else if (which == 2) mm_tile<0, C_, C_>(a1, pWv, m2, tile, lane);   // v
        else                 mm_tile<1, C_, C_>(a3, pWg, m3, tile, lane);   // g=silu
      }
      __syncthreads();

      // ---- phase 3: WKV recurrence + GroupNorm + *g  (state lives in LDS) ----
      if (wave < ROWS) {
        const int row = wave;
        const int c0 = lane * 2;                      // this lane: cols c0, c0+1
        const float v0 = m2[row * C_ + c0], v1 = m2[row * C_ + c0 + 1];
        float* sp = state + ((l * ROWS + row) * N_) * N_ + c0;
        float o0 = 0.f, o1 = 0.f;
#pragma unroll 4
        for (int i = 0; i < N_; ++i) {
          const float r = m0[row * C_ + i];
          const float k = m1[row * C_ + i];
          const float u = uL[l * N_ + i];
          const float w = wL[l * N_ + i];
          float2 s = *(float2*)(sp + i * N_);
          const float kv0 = k * v0, kv1 = k * v1;
          o0 += r * fmaf(u, kv0, s.x);
          o1 += r * fmaf(u, kv1, s.y);
          s.x = fmaf(w, s.x, kv0);
          s.y = fmaf(w, s.y, kv1);
          *(float2*)(sp + i * N_) = s;
        }
        // GroupNorm (H=1 group over 64 channels), then multiply by g -> a0
        const float mean = wred_sum(o0 + o1) * (1.f / 64.f);
        const float d0 = o0 - mean, d1 = o1 - mean;
        const float var = wred_sum(d0 * d0 + d1 * d1) * (1.f / 64.f);
        const float rs = rsqrtf(var + EPS_GN);
        const float g0 = (d0 * rs) * lnx_w[l * C_ + c0]     + lnx_b[l * C_ + c0];
        const float g1 = (d1 * rs) * lnx_w[l * C_ + c0 + 1] + lnx_b[l * C_ + c0 + 1];
        a0[row * C_ + c0]     = g0 * m3[row * C_ + c0];
        a0[row * C_ + c0 + 1] = g1 * m3[row * C_ + c0 + 1];
      }
      __syncthreads();

      // ---- phase 4: (o*g) @ Wo^T ----
      for (int job = wave; job < 4; job += NWAVES)
        mm_tile<0, C_, C_>(a0, pWo, m0, job, lane);
      __syncthreads();

      // ---- phase 5: h += att_out ; ln2 ; channel-mix -> a1(xk2), a2(xr2) ----
      if (wave < ROWS) {
        const int row = wave, i0 = row * C_ + lane;
        const float h0 = hbuf[i0] + m0[i0];
        const float h1 = hbuf[i0 + 32] + m0[i0 + 32];
        hbuf[i0] = h0; hbuf[i0 + 32] = h1;
        float n0, n1;
        ln_pair(h0, h1, EPS_LN,
                ln2_w[l * C_ + lane], ln2_w[l * C_ + lane + 32],
                ln2_b[l * C_ + lane], ln2_b[l * C_ + lane + 32], n0, n1);
        const int xi = (l * ROWS + row) * C_ + lane;
        const float p0 = ffnxx[xi], p1 = ffnxx[xi + 32];
        ffnxx[xi] = n0; ffnxx[xi + 32] = n1;
        const float fk0 = fmix_k[l * C_ + lane], fk1 = fmix_k[l * C_ + lane + 32];
        const float fr0 = fmix_r[l * C_ + lane], fr1 = fmix_r[l * C_ + lane + 32];
        a1[i0] = n0 * fk0 + p0 * (1.f - fk0);  a1[i0 + 32] = n1 * fk1 + p1 * (1.f - fk1);
        a2[i0] = n0 * fr0 + p0 * (1.f - fr0);  a2[i0 + 32] = n1 * fr1 + p1 * (1.f - fr1);
      }
      __syncthreads();

      // ---- phase 6: rr = sigmoid(xr2@fWr^T) ; kk = relu(xk2@fWk^T)^2 ----
      for (int job = wave; job < 20; job += NWAVES) {
        if (job < 4) mm_tile<2, C_, C_>(a2, pfWr, m3, job, lane);        // rr
        else         mm_tile<3, C_, F_>(a1, pfWk, aK, job - 4, lane);    // kk
      }
      __syncthreads();

      // ---- phase 7: kk @ fWv^T  (K = 256) ----
      for (int job = wave; job < 4; job += NWAVES)
        mm_tile<0, F_, C_>(aK, pfWv, m1, job, lane);
      __syncthreads();

      // ---- phase 8: h += rr * ffn_out ----
      if (wave < ROWS) {
        const int row = wave, i0 = row * C_ + lane;
        hbuf[i0]      += m3[i0]      * m1[i0];
        hbuf[i0 + 32] += m3[i0 + 32] * m1[i0 + 32];
      }
      __syncthreads();
    } // layers

    // ---- readout only needed on the final step: lno-LN then head ----
    if (t == T_ - 1) {
      if (wave < ROWS) {
        const int row = wave, i0 = row * C_ + lane;
        float n0, n1;
        ln_pair(hbuf[i0], hbuf[i0 + 32], EPS_LN,
                lno_w[lane], lno_w[lane + 32], lno_b[lane], lno_b[lane + 32],
                n0, n1);
        a3[i0] = n0; a3[i0 + 32] = n1;
      }
      __syncthreads();
      if (tid < ROWS * DOUT) {
        const int row = tid / DOUT, o = tid % DOUT;
        float acc = head_b[o];
#pragma unroll
        for (int c = 0; c < C_; ++c)
          acc = fmaf(a3[row * C_ + c], head_w[o * C_ + c], acc);
        out[(b0 + row) * DOUT + o] = acc;
      }
    }
    __syncthreads();
  } // time
}

// d_in flattening assumed: depth-first dict order of setup_inputs():
//  0:x 1:enc_w 2:enc_b 3:ln0_w 4:ln0_b 5:lno_w 6:lno_b 7:head_w 8:head_b
//  9:ln1_w 10:ln1_b 11:ln2_w 12:ln2_b 13:lnx_w 14:lnx_b
// 15:mix_k 16:mix_v 17:mix_r 18:mix_g 19:Wr 20:Wk 21:Wv 22:Wg 23:Wo
// 24:time_decay 25:time_first 26:fmix_k 27:fmix_r 28:fWk 29:fWv 30:fWr
extern "C" void kernel_launch(void* const* d_in, const int* in_sizes, int n_in,
                              void* d_out, int out_size, void* d_ws, size_t ws_size,
                              hipStream_t stream) {
  (void)in_sizes; (void)n_in; (void)out_size; (void)d_ws; (void)ws_size;
  const float* F[31];
  for (int i = 0; i < 31; ++i) F[i] = (const float*)d_in[i];

  (void)hipFuncSetAttribute((const void*)rwkv5_forward,
                            hipFuncAttributeMaxDynamicSharedMemorySize,
                            (int)SMEM_BYTES);

  rwkv5_forward<<<NBLK, NTHREADS, SMEM_BYTES, stream>>>(
      F[0], F[1], F[2], F[3], F[4], F[5], F[6], F[7], F[8],
      F[9], F[10], F[11], F[12], F[13], F[14],
      F[15], F[16], F[17], F[18],
      F[19], F[20], F[21], F[22], F[23],
      F[24], F[25], F[26], F[27],
      F[28], F[29], F[30],
      (float*)d_out);
}